// Network_73212012528347
// MI455X (gfx1250) — compile-verified
//
#include <hip/hip_runtime.h>
#include <hip/hip_bf16.h>

// ---------------------------------------------------------------------------
// Types for CDNA5 WMMA f32 16x16x4:  D(16x16,f32) = A(16x4,f32)*B(4x16,f32)+C
// A/B fragments: 64 f32 / 32 lanes = 2 VGPRs -> float2 ext vector
// C/D fragment : 256 f32 / 32 lanes = 8 VGPRs -> float8 ext vector
// ---------------------------------------------------------------------------
typedef __attribute__((ext_vector_type(2))) float v2f;
typedef __attribute__((ext_vector_type(4))) float v4f;
typedef __attribute__((ext_vector_type(8))) float v8f;

#define NSAMP   2000
#define CPH     100352      // 2048*7*7 per-sample elements
#define TOPK    5

// workspace layout (float offsets), all multiples of 4 -> 16B aligned
#define OFF_SCORES 0
#define OFF_TOPW   2000
#define OFF_TOPI   2008
#define OFF_POOLED 2016                       // 1024*14*14 = 200704
#define OFF_A1     (OFF_POOLED + 200704)      // 49*4096    = 200704
#define OFF_XBUF   (OFF_A1 + 200704)          // 6*49*2048  = 602112
#define OFF_Y      (OFF_XBUF + 602112)        // 294*512    = 150528
#define OFF_MEANS  (OFF_Y + 150528)           // 3072
#define OFF_FC1O   (OFF_MEANS + 3072)         // 512

// ---------------------------------------------------------------------------
// K1: scores[n] = dot(bottleneck[n], sel_w) + sel_b
// Dominant cost: ~803MB streamed once. Bottleneck (803MB) >> L2 (192MB),
// so stream it with non-temporal loads; sel_w (392KB) stays regular so it
// remains L2-resident across all 2000 blocks.
// ---------------------------------------------------------------------------
__global__ __launch_bounds__(256) void scores_kernel(
    const float* __restrict__ bot, const float* __restrict__ selw,
    const float* __restrict__ selb, float* __restrict__ scores)
{
    const int n = blockIdx.x;
    const v4f* a = (const v4f*)(bot + (size_t)n * CPH);
    const v4f* w = (const v4f*)selw;
    float s = 0.f;
    for (int i = threadIdx.x; i < CPH / 4; i += 256) {
        v4f x = __builtin_nontemporal_load(a + i);   // TH_NT: bypass-ish L2
        v4f y = w[i];                                 // regular: keep cached
        s += x.x * y.x + x.y * y.y + x.z * y.z + x.w * y.w;
    }
    __shared__ float red[256];
    red[threadIdx.x] = s;
    __syncthreads();
    for (int off = 128; off > 0; off >>= 1) {
        if (threadIdx.x < off) red[threadIdx.x] += red[threadIdx.x + off];
        __syncthreads();
    }
    if (threadIdx.x == 0) scores[n] = red[0] + selb[0];
}

// ---------------------------------------------------------------------------
// K2: top-5 (stable argmax, ties -> lower index) + softmax over ALL scores
// ---------------------------------------------------------------------------
__global__ __launch_bounds__(256) void topk_kernel(
    const float* __restrict__ scores, int N,
    int* __restrict__ topi, float* __restrict__ topw)
{
    __shared__ float ls[2048];
    __shared__ float rv[256];
    __shared__ int   ri[256];
    __shared__ float selv[TOPK];
    __shared__ int   seli[TOPK];
    const float NEGINF = -__builtin_inff();

    for (int i = threadIdx.x; i < 2048; i += 256)
        ls[i] = (i < N) ? scores[i] : NEGINF;
    __syncthreads();

    for (int r = 0; r < TOPK; ++r) {
        float bv = NEGINF; int bi = 0x7fffffff;
        for (int i = threadIdx.x; i < 2048; i += 256) {
            float v = ls[i];
            if (v > bv || (v == bv && i < bi)) { bv = v; bi = i; }
        }
        rv[threadIdx.x] = bv; ri[threadIdx.x] = bi;
        __syncthreads();
        for (int off = 128; off > 0; off >>= 1) {
            if (threadIdx.x < off) {
                float v2 = rv[threadIdx.x + off]; int i2 = ri[threadIdx.x + off];
                if (v2 > rv[threadIdx.x] ||
                    (v2 == rv[threadIdx.x] && i2 < ri[threadIdx.x])) {
                    rv[threadIdx.x] = v2; ri[threadIdx.x] = i2;
                }
            }
            __syncthreads();
        }
        if (threadIdx.x == 0) {
            selv[r] = rv[0]; seli[r] = ri[0];
            ls[ri[0]] = NEGINF;            // remove for next round
        }
        __syncthreads();
    }

    // softmax over all N scores (removed entries contribute via selv)
    float mx = selv[0];
    float ps = 0.f;
    for (int i = threadIdx.x; i < 2048; i += 256) ps += expf(ls[i] - mx); // -inf -> 0
    rv[threadIdx.x] = ps;
    __syncthreads();
    for (int off = 128; off > 0; off >>= 1) {
        if (threadIdx.x < off) rv[threadIdx.x] += rv[threadIdx.x + off];
        __syncthreads();
    }
    if (threadIdx.x == 0) {
        float sum = rv[0];
        float e[TOPK];
        for (int j = 0; j < TOPK; ++j) { e[j] = expf(selv[j] - mx); sum += e[j]; }
        for (int j = 0; j < TOPK; ++j) { topw[j] = e[j] / sum; topi[j] = seli[j]; }
    }
}

// ---------------------------------------------------------------------------
// K3: 4x4 average pool (1,1024,56,56) -> (1024,14,14)
// ---------------------------------------------------------------------------
__global__ __launch_bounds__(256) void pool_kernel(
    const float* __restrict__ gin, float* __restrict__ pooled)
{
    int e = blockIdx.x * 256 + threadIdx.x;       // 1024*196
    if (e >= 1024 * 196) return;
    int c = e / 196, r = e % 196, y = r / 14, x = r % 14;
    const float* p = gin + (size_t)c * 3136 + (size_t)y * 4 * 56 + x * 4;
    float s = 0.f;
    #pragma unroll
    for (int i = 0; i < 4; ++i)
        #pragma unroll
        for (int j = 0; j < 4; ++j) s += p[i * 56 + j];
    pooled[e] = s * (1.f / 16.f);
}

// ---------------------------------------------------------------------------
// K4: im2col for 2x2/stride2 conv: A1[p][ic*4+ky*2+kx] = pooled[ic][2oy+ky][2ox+kx]
// ---------------------------------------------------------------------------
__global__ __launch_bounds__(256) void im2col_kernel(
    const float* __restrict__ pooled, float* __restrict__ A1)
{
    int e = blockIdx.x * 256 + threadIdx.x;       // 49*4096
    if (e >= 49 * 4096) return;
    int p = e >> 12, k = e & 4095;
    int ic = k >> 2, ky = (k >> 1) & 1, kx = k & 1;
    int oy = p / 7, ox = p % 7;
    A1[e] = pooled[ic * 196 + (2 * oy + ky) * 14 + (2 * ox + kx)];
}

// ---------------------------------------------------------------------------
// K6: gather top-5 samples scaled by softmax weights into [j][p][c] layout
// ---------------------------------------------------------------------------
__global__ __launch_bounds__(256) void gather_kernel(
    const float* __restrict__ bot, const int* __restrict__ topi,
    const float* __restrict__ topw, float* __restrict__ xbuf)
{
    int e = blockIdx.x * 256 + threadIdx.x;       // 5*49*2048
    if (e >= TOPK * 49 * 2048) return;
    int c = e & 2047;
    int t = e >> 11;                              // j*49 + p
    int j = t / 49, p = t % 49;
    xbuf[e] = bot[(size_t)topi[j] * CPH + (size_t)c * 49 + p] * topw[j];
}

// ---------------------------------------------------------------------------
// WMMA f32 GEMM: C[M,N] = PReLU(A[MxK] * B[NxK]^T + bias)  (row-major A,B)
// 256 threads = 8 waves as 2(M) x 4(N); block tile 32x64; K chunked by 64.
// Double-buffered LDS (ping-pong): prefetch next chunk's global data into
// registers before the 16-WMMA burst, one barrier per chunk. LDS pitch 68
// floats -> conflict-free column-strided fragment reads.
// ---------------------------------------------------------------------------
#define KC 64
#define PITCH 68
__global__ __launch_bounds__(256) void gemm_wmma_f32(
    const float* __restrict__ A, const float* __restrict__ B,
    const float* __restrict__ bias, const float* __restrict__ alpha,
    float* __restrict__ C, int M, int N, int K, int ldc)
{
    __shared__ float As[2][32][PITCH];
    __shared__ float Bs[2][64][PITCH];
    const int tid  = threadIdx.x;
    const int lane = tid & 31;
    const int wave = tid >> 5;
    const int wm   = wave & 1;        // 0..1  (M tile)
    const int wn   = wave >> 1;       // 0..3  (N tile)
    const int bM0  = blockIdx.x * 32;
    const int bN0  = blockIdx.y * 64;

    // cooperative-load coordinates (fixed per thread)
    const int arow0 = (tid) >> 4,          ac4 = tid & 15;          // A it=0
    const int arow1 = (tid + 256) >> 4;                             // A it=1
    const int brow0 = tid >> 4;                                     // B it: row0 + it*16

    v8f acc = {};
    const int mrow = wm * 16 + (lane & 15);
    const int nrow = wn * 16 + (lane & 15);
    const int half = (lane >> 4) << 1;    // lanes 0-15 -> K{0,1}; 16-31 -> K{2,3}

    v4f aReg[2], bReg[4];
    const v4f zero = {0.f, 0.f, 0.f, 0.f};

    // ---- prefetch chunk 0 into registers ----
    {
        int gr0 = bM0 + arow0, gr1 = bM0 + arow1;
        aReg[0] = (gr0 < M) ? *(const v4f*)(A + (size_t)gr0 * K + ac4 * 4) : zero;
        aReg[1] = (gr1 < M) ? *(const v4f*)(A + (size_t)gr1 * K + ac4 * 4) : zero;
        #pragma unroll
        for (int it = 0; it < 4; ++it)
            bReg[it] = *(const v4f*)(B + (size_t)(bN0 + brow0 + it * 16) * K + ac4 * 4);
    }
    // ---- store chunk 0 into buffer 0 ----
    *(v4f*)(&As[0][arow0][ac4 * 4]) = aReg[0];
    *(v4f*)(&As[0][arow1][ac4 * 4]) = aReg[1];
    #pragma unroll
    for (int it = 0; it < 4; ++it)
        *(v4f*)(&Bs[0][brow0 + it * 16][ac4 * 4]) = bReg[it];
    __syncthreads();

    int cur = 0;
    for (int k0 = 0; k0 < K; k0 += KC) {
        const bool hasNext = (k0 + KC) < K;
        // ---- prefetch next chunk (global -> regs), overlapped with WMMA ----
        if (hasNext) {
            int kn = k0 + KC;
            int gr0 = bM0 + arow0, gr1 = bM0 + arow1;
            aReg[0] = (gr0 < M) ? *(const v4f*)(A + (size_t)gr0 * K + kn + ac4 * 4) : zero;
            aReg[1] = (gr1 < M) ? *(const v4f*)(A + (size_t)gr1 * K + kn + ac4 * 4) : zero;
            #pragma unroll
            for (int it = 0; it < 4; ++it)
                bReg[it] = *(const v4f*)(B + (size_t)(bN0 + brow0 + it * 16) * K + kn + ac4 * 4);
        }
        // ---- 16 WMMAs on current buffer ----
        #pragma unroll
        for (int kk = 0; kk < KC; kk += 4) {
            v2f a, b;
            a.x = As[cur][mrow][kk + half];
            a.y = As[cur][mrow][kk + half + 1];
            b.x = Bs[cur][nrow][kk + half];
            b.y = Bs[cur][nrow][kk + half + 1];
            acc = __builtin_amdgcn_wmma_f32_16x16x4_f32(
                false, a, false, b, (short)0, acc, false, false);
        }
        // ---- commit prefetched chunk into the other buffer ----
        if (hasNext) {
            int nxt = cur ^ 1;
            *(v4f*)(&As[nxt][arow0][ac4 * 4]) = aReg[0];
            *(v4f*)(&As[nxt][arow1][ac4 * 4]) = aReg[1];
            #pragma unroll
            for (int it = 0; it < 4; ++it)
                *(v4f*)(&Bs[nxt][brow0 + it * 16][ac4 * 4]) = bReg[it];
            __syncthreads();
            cur = nxt;
        }
    }

    // epilogue: C/D layout: VGPR r -> M = r (lanes 0-15) or r+8 (lanes 16-31)
    const int n  = bN0 + wn * 16 + (lane & 15);   // always < N (N % 64 == 0)
    const float bn = bias[n];
    const float an = alpha[n];
    const int mhi = (lane >> 4) << 3;
    #pragma unroll
    for (int r = 0; r < 8; ++r) {
        int m = bM0 + wm * 16 + r + mhi;
        if (m < M) {
            float v = acc[r] + bn;
            C[(size_t)m * ldc + n] = (v >= 0.f) ? v : an * v;
        }
    }
}

// ---------------------------------------------------------------------------
// K8: spatial mean over p=49 -> means[n*512+oc]
// ---------------------------------------------------------------------------
__global__ __launch_bounds__(256) void mean_kernel(
    const float* __restrict__ y, float* __restrict__ means)
{
    int e = blockIdx.x * 256 + threadIdx.x;       // 6*512
    if (e >= 3072) return;
    int n = e >> 9, oc = e & 511;
    const float* p = y + (size_t)n * 49 * 512 + oc;
    float s = 0.f;
    #pragma unroll
    for (int i = 0; i < 49; ++i) s += p[i * 512];
    means[e] = s * (1.f / 49.f);
}

// ---------------------------------------------------------------------------
// K9: fc1  out[o] = PReLU(dot(means[3072], fc1_w[o]) + b[o])
// ---------------------------------------------------------------------------
__global__ __launch_bounds__(128) void fc1_kernel(
    const float* __restrict__ x, const float* __restrict__ w,
    const float* __restrict__ b, const float* __restrict__ alpha,
    float* __restrict__ out)
{
    int o = blockIdx.x;
    const float* wr = w + (size_t)o * 3072;
    float s = 0.f;
    for (int i = threadIdx.x; i < 3072; i += 128) s += x[i] * wr[i];
    __shared__ float red[128];
    red[threadIdx.x] = s;
    __syncthreads();
    for (int off = 64; off > 0; off >>= 1) {
        if (threadIdx.x < off) red[threadIdx.x] += red[threadIdx.x + off];
        __syncthreads();
    }
    if (threadIdx.x == 0) {
        float v = red[0] + b[o];
        out[o] = (v >= 0.f) ? v : alpha[o] * v;
    }
}

// ---------------------------------------------------------------------------
// K10: fc2  out[o] = dot(fc1o[512], fc2_w[o]) + b[o]
// ---------------------------------------------------------------------------
__global__ __launch_bounds__(128) void fc2_kernel(
    const float* __restrict__ x, const float* __restrict__ w,
    const float* __restrict__ b, float* __restrict__ out)
{
    int o = blockIdx.x;
    const float* wr = w + (size_t)o * 512;
    float s = 0.f;
    for (int i = threadIdx.x; i < 512; i += 128) s += x[i] * wr[i];
    __shared__ float red[128];
    red[threadIdx.x] = s;
    __syncthreads();
    for (int off = 64; off > 0; off >>= 1) {
        if (threadIdx.x < off) red[threadIdx.x] += red[threadIdx.x + off];
        __syncthreads();
    }
    if (threadIdx.x == 0) out[o] = red[0] + b[o];
}

// ---------------------------------------------------------------------------
extern "C" void kernel_launch(void* const* d_in, const int* in_sizes, int n_in,
                              void* d_out, int out_size, void* d_ws, size_t ws_size,
                              hipStream_t stream)
{
    (void)in_sizes; (void)n_in; (void)out_size; (void)ws_size;

    const float* bot     = (const float*)d_in[0];   // (2000,2048,7,7)
    const float* gin     = (const float*)d_in[1];   // (1,1024,56,56)
    const float* selw    = (const float*)d_in[2];   // (1,2048,7,7)
    const float* selb    = (const float*)d_in[3];   // (1,)
    const float* globw   = (const float*)d_in[4];   // (2048,1024,2,2)
    const float* globb   = (const float*)d_in[5];   // (2048,)
    const float* globalp = (const float*)d_in[6];   // (2048,)
    const float* conv1w  = (const float*)d_in[7];   // (512,2048,1,1)
    const float* conv1b  = (const float*)d_in[8];   // (512,)
    const float* alpha1  = (const float*)d_in[9];   // (512,)
    const float* fc1w    = (const float*)d_in[10];  // (512,3072)
    const float* fc1b    = (const float*)d_in[11];  // (512,)
    const float* fc1a    = (const float*)d_in[12];  // (512,)
    const float* fc2w    = (const float*)d_in[13];  // (5,512)
    const float* fc2b    = (const float*)d_in[14];  // (5,)

    float* ws     = (float*)d_ws;
    float* scores = ws + OFF_SCORES;
    float* topw   = ws + OFF_TOPW;
    int*   topi   = (int*)(ws + OFF_TOPI);
    float* pooled = ws + OFF_POOLED;
    float* A1     = ws + OFF_A1;
    float* xbuf   = ws + OFF_XBUF;                  // [6][49][2048]
    float* ybuf   = ws + OFF_Y;                     // [294][512]
    float* means  = ws + OFF_MEANS;
    float* fc1o   = ws + OFF_FC1O;

    // 1) selection scores (dominant: ~803MB streamed, NT loads)
    scores_kernel<<<NSAMP, 256, 0, stream>>>(bot, selw, selb, scores);
    // 2) top-5 + softmax weights
    topk_kernel<<<1, 256, 0, stream>>>(scores, NSAMP, topi, topw);
    // 3) 4x4 avg pool
    pool_kernel<<<(1024 * 196 + 255) / 256, 256, 0, stream>>>(gin, pooled);
    // 4) im2col for the 2x2/stride2 conv
    im2col_kernel<<<(49 * 4096 + 255) / 256, 256, 0, stream>>>(pooled, A1);
    // 5) glob conv as WMMA GEMM: M=49, N=2048, K=4096 -> xbuf slot n=5 ([p][oc])
    gemm_wmma_f32<<<dim3(2, 32), 256, 0, stream>>>(
        A1, globw, globb, globalp, xbuf + (size_t)5 * 49 * 2048, 49, 2048, 4096, 2048);
    // 6) gather+scale top-5 samples into xbuf[0..4] ([p][c] layout)
    gather_kernel<<<(TOPK * 49 * 2048 + 255) / 256, 256, 0, stream>>>(bot, topi, topw, xbuf);
    // 7) conv1 (1x1) as WMMA GEMM: M=294, N=512, K=2048
    gemm_wmma_f32<<<dim3(10, 8), 256, 0, stream>>>(
        xbuf, conv1w, conv1b, alpha1, ybuf, 294, 512, 2048, 512);
    // 8) spatial mean -> (1, 3072)
    mean_kernel<<<12, 256, 0, stream>>>(ybuf, means);
    // 9) fc1 + PReLU
    fc1_kernel<<<512, 128, 0, stream>>>(means, fc1w, fc1b, fc1a, fc1o);
    // 10) fc2 -> d_out (5 floats)
    fc2_kernel<<<TOPK, 128, 0, stream>>>(fc1o, fc2w, fc2b, (float*)d_out);
}